// GaussianHFCFilter_9732395893273
// MI455X (gfx1250) — compile-verified
//
#include <hip/hip_runtime.h>
#include <hip/hip_bf16.h>
#include <math.h>

#define H_    512
#define W_    512
#define HW_   262144
#define CCH   3
#define NC_   48
#define XPS   153   // LDS row stride for hconv staging (coprime with 64 banks)

typedef __attribute__((ext_vector_type(2))) float v2f;
typedef __attribute__((ext_vector_type(8))) float v8f;

// async-to-LDS builtin operand types: vector_size(16) int, AS1 src / AS3 dst
typedef int v4i __attribute__((vector_size(16)));
typedef __attribute__((address_space(1))) v4i glob_v4i;
typedef __attribute__((address_space(3))) v4i lds_v4i;

__device__ __forceinline__ uint32_t mono_u(float f) {
    uint32_t u = __float_as_uint(f);
    return (u & 0x80000000u) ? ~u : (u | 0x80000000u);
}
__device__ __forceinline__ float unmono_f(uint32_t m) {
    uint32_t u = (m & 0x80000000u) ? (m & 0x7FFFFFFFu) : ~m;
    return __uint_as_float(u);
}

// ---------------------------------------------------------------------------
// Stage 0: separable 1D Gaussian taps from the 2D kernel (k2 = outer(k1,k1))
// w1[t] = k2[11][t] / sqrt(k2[11][11])
// ---------------------------------------------------------------------------
__global__ void __launch_bounds__(32) w1_kernel(const float* __restrict__ kern,
                                                float* __restrict__ w1) {
    int t = threadIdx.x;
    if (t < 23) {
        float s = sqrtf(kern[11 * 23 + 11]);
        w1[t] = kern[11 * 23 + t] / s;
    }
}

// ---------------------------------------------------------------------------
// Stage 1: exact per-channel median via 4-pass radix select (8 bits/pass)
// rank = (HW-1)/2 = 131071 ; med = value + 0.2
// ---------------------------------------------------------------------------
__global__ void __launch_bounds__(256) median_kernel(const float* __restrict__ x,
                                                     float* __restrict__ med) {
    int ch = blockIdx.x;
    const float* p = x + (size_t)ch * HW_;
    __shared__ uint32_t hist[256];
    __shared__ uint32_t s_prefix, s_want;
    int tid = threadIdx.x;
    if (tid == 0) { s_prefix = 0u; s_want = 131071u; }
    __syncthreads();
    for (int pass = 0; pass < 4; ++pass) {
        int shift = 24 - 8 * pass;
        hist[tid] = 0u;
        __syncthreads();
        uint32_t pref = s_prefix;
        for (int i = tid; i < HW_; i += 256) {
            uint32_t u = mono_u(p[i]);
            if (pass == 0 || ((u >> (shift + 8)) == pref))
                atomicAdd(&hist[(u >> shift) & 255u], 1u);
        }
        __syncthreads();
        if (tid == 0) {
            uint32_t want = s_want, cum = 0u;
            for (int b = 0; b < 256; ++b) {
                uint32_t h = hist[b];
                if (cum + h > want) { s_prefix = (pref << 8) | (uint32_t)b; s_want = want - cum; break; }
                cum += h;
            }
        }
        __syncthreads();
    }
    if (tid == 0) med[ch] = unmono_f(s_prefix) + 0.2f;
}

// Branchless banded tap: w1[t] if 0<=t<23 else 0 (unconditional LDS load)
__device__ __forceinline__ float band_tap(const float* w1s, int t) {
    int tc = t < 0 ? 0 : (t > 22 ? 22 : t);
    float v = w1s[tc];
    return (t == tc) ? v : 0.0f;
}

// ---------------------------------------------------------------------------
// Stage 2: horizontal 23-tap conv of xp = mask*x + (1-mask)*med, edge-clamped.
// Per wave: 16x16 output tile as 10 chained V_WMMA_F32_16X16X4_F32:
//   D[16x16] += A[16x4] (xp window from LDS) x B[4x16] (banded taps)
// Block = 8 waves -> 16 rows x 128 cols; LDS window 16 x 150 (stride 153).
// ---------------------------------------------------------------------------
__global__ void __launch_bounds__(256) hconv_kernel(const float* __restrict__ x,
                                                    const float* __restrict__ mask,
                                                    const float* __restrict__ med,
                                                    const float* __restrict__ w1g,
                                                    float* __restrict__ htmp) {
    __shared__ float xpl[16 * XPS];
    __shared__ float w1s[24];
    int b   = blockIdx.x;
    int ch  = b >> 7;          // 128 tiles per channel
    int rem = b & 127;
    int rt  = rem >> 2;        // 32 row tiles
    int ct  = rem & 3;         // 4 col tiles of 128
    int row0 = rt * 16, c0 = ct * 128;
    int tid = threadIdx.x;
    if (tid < 23) w1s[tid] = w1g[tid];
    float mv = med[ch];
    size_t xbase = (size_t)ch * HW_;
    size_t mbase = (size_t)(ch / CCH) * HW_;
    for (int idx = tid; idx < 16 * 150; idx += 256) {
        int m = idx / 150, j = idx % 150;
        int col = c0 - 11 + j;
        col = col < 0 ? 0 : (col > (W_ - 1) ? (W_ - 1) : col);
        int off = (row0 + m) * W_ + col;
        float xv = x[xbase + off];
        float mk = mask[mbase + off];
        xpl[m * XPS + j] = mk * xv + (1.0f - mk) * mv;
    }
    __syncthreads();

    int wv = tid >> 5, lane = tid & 31;
    int nn = lane & 15;
    bool hi16 = lane >= 16;
    // Banded B: B[J][n] = (0 <= J-n <= 22) ? w1[J-n] : 0  (K chunks of 4)
    float b0[10], b1[10];
#pragma unroll
    for (int k = 0; k < 10; ++k) {
        int J = 4 * k + (hi16 ? 2 : 0);
        b0[k] = band_tap(w1s, J - nn);
        b1[k] = band_tap(w1s, J + 1 - nn);
    }
    v8f acc = {};
    int colbase = 16 * wv;
#pragma unroll
    for (int k = 0; k < 10; ++k) {
        int J = 4 * k + (hi16 ? 2 : 0);
        v2f a, bb;
        a.x  = xpl[nn * XPS + colbase + J];
        a.y  = xpl[nn * XPS + colbase + J + 1];
        bb.x = b0[k];
        bb.y = b1[k];
        acc = __builtin_amdgcn_wmma_f32_16x16x4_f32(false, a, false, bb,
                                                    (short)0, acc, false, false);
    }
    int outc = c0 + 16 * wv + nn;
#pragma unroll
    for (int i = 0; i < 8; ++i) {
        int rr = row0 + i + (hi16 ? 8 : 0);
        htmp[xbase + (size_t)rr * W_ + outc] = acc[i];
    }
}

// ---------------------------------------------------------------------------
// Stage 3: vertical 23-tap conv of htmp (edge-clamped rows) + res fusion:
//   res = 4*(xp - gauss).  Per wave 16x16 tile, A = banded taps, B = htmp rows.
// Block = 8 waves -> 128 rows x 16 cols; LDS window 150 x 16, staged with
// GLOBAL_LOAD_ASYNC_TO_LDS_B128 (ASYNCcnt) — no VGPR round trip.
// ---------------------------------------------------------------------------
__global__ void __launch_bounds__(256) vconv_kernel(const float* __restrict__ x,
                                                    const float* __restrict__ mask,
                                                    const float* __restrict__ med,
                                                    const float* __restrict__ w1g,
                                                    const float* __restrict__ htmp,
                                                    float* __restrict__ resb) {
    __shared__ float hl[150 * 16];
    __shared__ float w1s[24];
    int b   = blockIdx.x;
    int ch  = b >> 7;
    int rem = b & 127;
    int ct  = rem >> 2;        // 32 col tiles of 16
    int rt  = rem & 3;         // 4 row tiles of 128
    int c0 = ct * 16, r0 = rt * 128;
    int tid = threadIdx.x;
    if (tid < 23) w1s[tid] = w1g[tid];
    size_t cbase = (size_t)ch * HW_;
    // Async copy: 150 rows x 16 floats = 600 B128 transfers, direct to LDS.
    for (int idx = tid; idx < 150 * 4; idx += 256) {
        int j = idx >> 2, q = idx & 3;
        int row = r0 - 11 + j;
        row = row < 0 ? 0 : (row > (H_ - 1) ? (H_ - 1) : row);
        const float* src = htmp + cbase + (size_t)row * W_ + c0 + q * 4;
        float* dst = hl + j * 16 + q * 4;
        __builtin_amdgcn_global_load_async_to_lds_b128((glob_v4i*)src, (lds_v4i*)dst,
                                                       0, 0);
    }
    asm volatile("s_wait_asynccnt 0x0" ::: "memory");
    __syncthreads();

    int wv = tid >> 5, lane = tid & 31;
    int nn = lane & 15;
    bool hi16 = lane >= 16;
    // Banded A: A[m][J] = (0 <= J-m <= 22) ? w1[J-m] : 0
    float a0[10], a1[10];
#pragma unroll
    for (int k = 0; k < 10; ++k) {
        int J = 4 * k + (hi16 ? 2 : 0);
        a0[k] = band_tap(w1s, J - nn);
        a1[k] = band_tap(w1s, J + 1 - nn);
    }
    v8f acc = {};
    int jbase = 16 * wv;
#pragma unroll
    for (int k = 0; k < 10; ++k) {
        int J = jbase + 4 * k + (hi16 ? 2 : 0);
        v2f a, bb;
        a.x  = a0[k];
        a.y  = a1[k];
        bb.x = hl[J * 16 + nn];
        bb.y = hl[(J + 1) * 16 + nn];
        acc = __builtin_amdgcn_wmma_f32_16x16x4_f32(false, a, false, bb,
                                                    (short)0, acc, false, false);
    }
    size_t mbase = (size_t)(ch / CCH) * HW_;
    float mv = med[ch];
    int cc = c0 + nn;
#pragma unroll
    for (int i = 0; i < 8; ++i) {
        int rr = r0 + 16 * wv + i + (hi16 ? 8 : 0);
        size_t off = (size_t)rr * W_ + cc;
        float xv = x[cbase + off];
        float mk = mask[mbase + off];
        float xp = mk * xv + (1.0f - mk) * mv;
        resb[cbase + off] = 4.0f * (xp - acc[i]);
    }
}

// ---------------------------------------------------------------------------
// Stage 4: exact 3%/97% percentiles of temp = trunc(res*256)/256.
// Resolve 4 ranks (7864,7865,254278,254279) simultaneously: 4 histograms,
// 4 passes over the channel. lo = v0 + .29(v1-v0); hi = v2 + .71(v3-v2).
// ---------------------------------------------------------------------------
__global__ void __launch_bounds__(256) pct_kernel(const float* __restrict__ resb,
                                                  float* __restrict__ lo,
                                                  float* __restrict__ hi) {
    int ch = blockIdx.x;
    const float* p = resb + (size_t)ch * HW_;
    __shared__ uint32_t hist[4][256];
    __shared__ uint32_t s_pref[4], s_want[4];
    int tid = threadIdx.x;
    if (tid < 4) s_pref[tid] = 0u;
    if (tid == 0) {
        s_want[0] = 7864u;   s_want[1] = 7865u;    // 0.03*(HW-1) = 7864.29
        s_want[2] = 254278u; s_want[3] = 254279u;  // 0.97*(HW-1) = 254278.71
    }
    __syncthreads();
    for (int pass = 0; pass < 4; ++pass) {
        int shift = 24 - 8 * pass;
        for (int r = 0; r < 4; ++r) hist[r][tid] = 0u;
        __syncthreads();
        uint32_t pr0 = s_pref[0], pr1 = s_pref[1], pr2 = s_pref[2], pr3 = s_pref[3];
        for (int i = tid; i < HW_; i += 256) {
            float t = truncf(p[i] * 256.0f) * (1.0f / 256.0f);
            uint32_t u   = mono_u(t);
            uint32_t bin = (u >> shift) & 255u;
            uint32_t up  = (pass == 0) ? 0u : (u >> (shift + 8));
            if (pass == 0 || up == pr0) atomicAdd(&hist[0][bin], 1u);
            if (pass == 0 || up == pr1) atomicAdd(&hist[1][bin], 1u);
            if (pass == 0 || up == pr2) atomicAdd(&hist[2][bin], 1u);
            if (pass == 0 || up == pr3) atomicAdd(&hist[3][bin], 1u);
        }
        __syncthreads();
        if (tid < 4) {
            uint32_t want = s_want[tid], cum = 0u, pref = s_pref[tid];
            for (int b = 0; b < 256; ++b) {
                uint32_t h = hist[tid][b];
                if (cum + h > want) { s_pref[tid] = (pref << 8) | (uint32_t)b; s_want[tid] = want - cum; break; }
                cum += h;
            }
        }
        __syncthreads();
    }
    if (tid == 0) {
        float v0 = unmono_f(s_pref[0]), v1 = unmono_f(s_pref[1]);
        float v2 = unmono_f(s_pref[2]), v3 = unmono_f(s_pref[3]);
        lo[ch] = v0 + 0.29f * (v1 - v0);
        hi[ch] = v2 + 0.71f * (v3 - v2);
    }
}

// ---------------------------------------------------------------------------
// Stage 5: out = (res - lo)/(hi - lo) * mask   (float4 vectorized)
// ---------------------------------------------------------------------------
__global__ void __launch_bounds__(256) final_kernel(const float* __restrict__ resb,
                                                    const float* __restrict__ mask,
                                                    const float* __restrict__ lo,
                                                    const float* __restrict__ hi,
                                                    float* __restrict__ out) {
    size_t g = ((size_t)blockIdx.x * blockDim.x + threadIdx.x) * 4;
    if (g >= (size_t)NC_ * HW_) return;
    int ch = (int)(g / HW_);
    size_t inoff = g % HW_;
    const float4 r  = *(const float4*)(resb + g);
    const float4 mk = *(const float4*)(mask + (size_t)(ch / CCH) * HW_ + inoff);
    float l = lo[ch], h = hi[ch];
    float inv = 1.0f / (h - l);
    float4 o;
    o.x = (r.x - l) * inv * mk.x;
    o.y = (r.y - l) * inv * mk.y;
    o.z = (r.z - l) * inv * mk.z;
    o.w = (r.w - l) * inv * mk.w;
    *(float4*)(out + g) = o;
}

// ---------------------------------------------------------------------------
extern "C" void kernel_launch(void* const* d_in, const int* in_sizes, int n_in,
                              void* d_out, int out_size, void* d_ws, size_t ws_size,
                              hipStream_t stream) {
    const float* x    = (const float*)d_in[0];
    const float* mask = (const float*)d_in[1];
    const float* kern = (const float*)d_in[2];
    float* out = (float*)d_out;

    float* ws   = (float*)d_ws;
    float* htmp = ws;                                // NC*HW floats
    float* resb = htmp + (size_t)NC_ * HW_;          // NC*HW floats
    float* med  = resb + (size_t)NC_ * HW_;          // 64 floats
    float* lo   = med + 64;                          // 64 floats
    float* hi   = lo + 64;                           // 64 floats
    float* w1   = hi + 64;                           // 23 floats

    w1_kernel<<<1, 32, 0, stream>>>(kern, w1);
    median_kernel<<<NC_, 256, 0, stream>>>(x, med);
    hconv_kernel<<<NC_ * 128, 256, 0, stream>>>(x, mask, med, w1, htmp);
    vconv_kernel<<<NC_ * 128, 256, 0, stream>>>(x, mask, med, w1, htmp, resb);
    pct_kernel<<<NC_, 256, 0, stream>>>(resb, lo, hi);
    final_kernel<<<(NC_ * HW_ / 4 + 255) / 256, 256, 0, stream>>>(resb, mask, lo, hi, out);
}